// CurvatureRegularization_20246475833445
// MI455X (gfx1250) — compile-verified
//
#include <hip/hip_runtime.h>
#include <hip/hip_bf16.h>
#include <stdint.h>

// ---------------------------------------------------------------------------
// CurvatureRegularization for MI455X (gfx1250, wave32)
//   pass 1: init nd[]=0, accum=0; compact phi (2MB) + pack pos->float4 (8MB)
//   pass 2: stream edge_index with NT B128 loads, gather phi/pos4 from L2,
//           two global_atomic_add_f32 per edge into interleaved float2 nd[]
//   pass 3: curvature^2 reduce; wave-level sum via V_WMMA_F32_16X16X4_F32
//   pass 4: scale + write scalar
// ---------------------------------------------------------------------------

typedef float v2f  __attribute__((ext_vector_type(2)));
typedef float v8f  __attribute__((ext_vector_type(8)));
typedef int   v4i  __attribute__((ext_vector_type(4)));

#define EPSF   1e-8f
#define WEIGHTF 0.01f
#define BLK 256

// ---- Pass 1a: init + compact/pack (packed path) ---------------------------
__global__ void k_init_pack(const float* __restrict__ x,
                            const float* __restrict__ pos,
                            float2* __restrict__ nd,
                            float*  __restrict__ phi,
                            float4* __restrict__ pos4,
                            float*  __restrict__ accum,
                            int N) {
  int i = blockIdx.x * blockDim.x + threadIdx.x;
  if (i == 0) *accum = 0.0f;
  if (i >= N) return;
  nd[i] = make_float2(0.0f, 0.0f);
  // streaming reads: non-temporal, do not pollute L2
  phi[i] = __builtin_nontemporal_load(x + (size_t)i * 16 + 8);
  size_t p3 = (size_t)i * 3;
  float px = __builtin_nontemporal_load(pos + p3 + 0);
  float py = __builtin_nontemporal_load(pos + p3 + 1);
  float pz = __builtin_nontemporal_load(pos + p3 + 2);
  pos4[i] = make_float4(px, py, pz, 0.0f);
}

// ---- Pass 1b: init only (fallback when ws too small for packing) ----------
__global__ void k_init_only(float2* __restrict__ nd,
                            float*  __restrict__ accum,
                            int N) {
  int i = blockIdx.x * blockDim.x + threadIdx.x;
  if (i == 0) *accum = 0.0f;
  if (i >= N) return;
  nd[i] = make_float2(0.0f, 0.0f);
}

// ---- Edge body ------------------------------------------------------------
template<bool PACKED>
__device__ __forceinline__ void process_edge(int s, int d,
                                             const float* __restrict__ phi,
                                             const float4* __restrict__ pos4,
                                             const float* __restrict__ x,
                                             const float* __restrict__ pos,
                                             float2* __restrict__ nd) {
  float phs, phd, dx, dy, dz;
  if (PACKED) {
    phs = phi[s];
    phd = phi[d];
    float4 a = pos4[s];
    float4 b = pos4[d];
    dx = b.x - a.x; dy = b.y - a.y; dz = b.z - a.z;
  } else {
    phs = x[(size_t)s * 16 + 8];
    phd = x[(size_t)d * 16 + 8];
    size_t s3 = (size_t)s * 3, d3 = (size_t)d * 3;
    dx = pos[d3 + 0] - pos[s3 + 0];
    dy = pos[d3 + 1] - pos[s3 + 1];
    dz = pos[d3 + 2] - pos[s3 + 2];
  }
  float d2 = fmaf(dx, dx, fmaf(dy, dy, fmaf(dz, dz, EPSF)));
  float c  = __fdividef(phd - phs, d2);
  atomicAdd(&nd[s].x, c);     // num
  atomicAdd(&nd[s].y, 1.0f);  // deg  (same cacheline as num)
}

// ---- Pass 2: edges, 4 per thread via NT B128 index loads ------------------
template<bool PACKED>
__global__ void k_edges_vec(const int* __restrict__ src,
                            const int* __restrict__ dst,
                            const float* __restrict__ phi,
                            const float4* __restrict__ pos4,
                            const float* __restrict__ x,
                            const float* __restrict__ pos,
                            float2* __restrict__ nd,
                            long long E) {
  long long t = (long long)blockIdx.x * blockDim.x + threadIdx.x;
  long long e = t * 4;
  if (e >= E) return;
  v4i s4 = __builtin_nontemporal_load((const v4i*)(src + e));
  v4i d4 = __builtin_nontemporal_load((const v4i*)(dst + e));
#pragma unroll
  for (int k = 0; k < 4; ++k)
    process_edge<PACKED>(s4[k], d4[k], phi, pos4, x, pos, nd);
}

// ---- Pass 2 fallback: one edge per thread (E not multiple of 4) -----------
template<bool PACKED>
__global__ void k_edges_scalar(const int* __restrict__ src,
                               const int* __restrict__ dst,
                               const float* __restrict__ phi,
                               const float4* __restrict__ pos4,
                               const float* __restrict__ x,
                               const float* __restrict__ pos,
                               float2* __restrict__ nd,
                               long long E) {
  long long e = (long long)blockIdx.x * blockDim.x + threadIdx.x;
  if (e >= E) return;
  process_edge<PACKED>(src[e], dst[e], phi, pos4, x, pos, nd);
}

// ---- Pass 3: curvature^2 reduction; wave sum via WMMA f32 16x16x4 ---------
// A(16x4) holds lane values (a0 = v, a1 = 0; lanes 0-15 -> K=0/1, 16-31 -> K=2/3
// per the 32-bit A layout). B = ones(4x16)  =>  D[m][n] = v_m + v_{m+16}.
// Per-lane sum of the 8 D VGPRs gives rows 0-7 (lanes 0-15) / 8-15 (lanes 16-31)
// partials; one shfl_xor(16) finishes the 32-lane total.
__global__ void k_finalize(const float2* __restrict__ nd,
                           float* __restrict__ accum,
                           int N) {
  int tid = threadIdx.x;
  float v = 0.0f;
  int stride = gridDim.x * blockDim.x;
  for (int i = blockIdx.x * blockDim.x + tid; i < N; i += stride) {
    float2 p = nd[i];
    float c = p.x / fmaxf(p.y, 1.0f);
    v = fmaf(c, c, v);
  }
  // ---- wave32 reduction through the matrix pipe (EXEC all 1s here) ----
  v2f a; a.x = v;    a.y = 0.0f;
  v2f b; b.x = 1.0f; b.y = 1.0f;
  v8f cz = {};
  v8f dm = __builtin_amdgcn_wmma_f32_16x16x4_f32(
      /*neg_a=*/false, a, /*neg_b=*/false, b,
      /*c_mod=*/(short)0, cz, /*reuse_a=*/false, /*reuse_b=*/false);
  float s = dm[0] + dm[1] + dm[2] + dm[3] + dm[4] + dm[5] + dm[6] + dm[7];
  s += __shfl_xor(s, 16, 32);

  __shared__ float wpart[BLK / 32];
  int lane = tid & 31, wid = tid >> 5;
  if (lane == 0) wpart[wid] = s;
  __syncthreads();
  if (tid == 0) {
    float tot = 0.0f;
#pragma unroll
    for (int w = 0; w < BLK / 32; ++w) tot += wpart[w];
    atomicAdd(accum, tot);
  }
}

// ---- Pass 4: scalar output ------------------------------------------------
__global__ void k_write(const float* __restrict__ accum,
                        float* __restrict__ out, float scale) {
  out[0] = accum[0] * scale;
}

// ---------------------------------------------------------------------------
extern "C" void kernel_launch(void* const* d_in, const int* in_sizes, int n_in,
                              void* d_out, int out_size, void* d_ws, size_t ws_size,
                              hipStream_t stream) {
  const float* x   = (const float*)d_in[0];
  const float* pos = (const float*)d_in[1];
  const int*   ei  = (const int*)d_in[2];

  const long long N = in_sizes[0] / 16;   // F = 16
  const long long E = in_sizes[2] / 2;    // edge_index is (2, E)
  const int* srcp = ei;
  const int* dstp = ei + E;

  auto alignup = [](uintptr_t v, uintptr_t al) { return (v + al - 1) & ~(al - 1); };
  uintptr_t base = (uintptr_t)d_ws;
  uintptr_t p = alignup(base, 16);
  float2* nd    = (float2*)p;  p = alignup(p + (size_t)N * sizeof(float2), 16);
  float*  accum = (float*)p;   p = alignup(p + sizeof(float), 16);
  float*  phi   = (float*)p;   p = alignup(p + (size_t)N * sizeof(float), 16);
  float4* pos4  = (float4*)p;  p += (size_t)N * sizeof(float4);
  bool packed = (ws_size >= (size_t)(p - base));

  long long initBlocks = (N + BLK - 1) / BLK;
  if (packed)
    k_init_pack<<<(int)initBlocks, BLK, 0, stream>>>(x, pos, nd, phi, pos4, accum, (int)N);
  else
    k_init_only<<<(int)initBlocks, BLK, 0, stream>>>(nd, accum, (int)N);

  if ((E & 3LL) == 0) {
    long long threads = E / 4;
    long long blocks  = (threads + BLK - 1) / BLK;
    if (packed)
      k_edges_vec<true ><<<(int)blocks, BLK, 0, stream>>>(srcp, dstp, phi, pos4, x, pos, nd, E);
    else
      k_edges_vec<false><<<(int)blocks, BLK, 0, stream>>>(srcp, dstp, phi, pos4, x, pos, nd, E);
  } else {
    long long blocks = (E + BLK - 1) / BLK;
    if (packed)
      k_edges_scalar<true ><<<(int)blocks, BLK, 0, stream>>>(srcp, dstp, phi, pos4, x, pos, nd, E);
    else
      k_edges_scalar<false><<<(int)blocks, BLK, 0, stream>>>(srcp, dstp, phi, pos4, x, pos, nd, E);
  }

  // ~4 elements per thread; modest atomic count into accum
  long long fblocks = (N + (long long)BLK * 4 - 1) / ((long long)BLK * 4);
  if (fblocks < 1) fblocks = 1;
  k_finalize<<<(int)fblocks, BLK, 0, stream>>>(nd, accum, (int)N);

  float scale = WEIGHTF / (float)N;
  k_write<<<1, 1, 0, stream>>>(accum, (float*)d_out, scale);
}